// GCN_66039417143825
// MI455X (gfx1250) — compile-verified
//
#include <hip/hip_runtime.h>
#include <hip/hip_bf16.h>

typedef float v2f __attribute__((ext_vector_type(2)));
typedef float v8f __attribute__((ext_vector_type(8)));

#define N_NODES  50000
#define N_EDGES  1600000
#define N_GRAPHS 64
#define FEAT     128
#define NCLS     32

// ---------------- degree histograms ----------------
__global__ void degrees_kernel(const int* __restrict__ src, const int* __restrict__ dst,
                               float* __restrict__ deg_out, float* __restrict__ deg_in, int E) {
    int e = blockIdx.x * blockDim.x + threadIdx.x;
    if (e < E) {
        atomicAdd(&deg_out[src[e]], 1.0f);
        atomicAdd(&deg_in[dst[e]], 1.0f);
    }
}

// ---------------- norms: rsqrt(max(deg,1)) ----------------
__global__ void norm_kernel(const float* __restrict__ deg_out, const float* __restrict__ deg_in,
                            float* __restrict__ norm_src, float* __restrict__ norm_dst, int N) {
    int i = blockIdx.x * blockDim.x + threadIdx.x;
    if (i < N) {
        norm_src[i] = rsqrtf(fmaxf(deg_out[i], 1.0f));
        norm_dst[i] = rsqrtf(fmaxf(deg_in[i], 1.0f));
    }
}

// ---------------- xs = x * norm_src (row scale, float4) ----------------
__global__ void scale_rows_kernel(const float* __restrict__ x, const float* __restrict__ s,
                                  float* __restrict__ y, int N) {
    int t = blockIdx.x * blockDim.x + threadIdx.x;   // N*32 threads
    int row = t >> 5;
    if (row >= N) return;
    int c = (t & 31) << 2;
    float sv = s[row];
    float4 v = *(const float4*)(x + (size_t)row * FEAT + c);
    v.x *= sv; v.y *= sv; v.z *= sv; v.w *= sv;
    *(float4*)(y + (size_t)row * FEAT + c) = v;
}

// ---------------- edge gather + scatter-add (one wave per edge) ----------------
__global__ void edge_agg_kernel(const float* __restrict__ xs, const int* __restrict__ src,
                                const int* __restrict__ dst, float* __restrict__ agg, int E) {
    int t = blockIdx.x * blockDim.x + threadIdx.x;
    int e = t >> 5;                      // 32 lanes cover one 128-float row
    if (e >= E) return;
    int c = (t & 31) << 2;
    int s = src[e], d = dst[e];
    float4 v = *(const float4*)(xs + (size_t)s * FEAT + c);
    float* o = agg + (size_t)d * FEAT + c;
    atomicAdd(o + 0, v.x);
    atomicAdd(o + 1, v.y);
    atomicAdd(o + 2, v.z);
    atomicAdd(o + 3, v.w);
}

// ---------------- fused GEMM: out = op((A*norm_dst) @ W + b) [* norm_src] ----------------
// One wave computes a 16x16 tile of out with V_WMMA_F32_16X16X4_F32 (full f32).
// Grid: N/16 blocks, blockDim = 32 * (DOUT/16) waves (one wave per column tile).
// 50000 % 16 == 0 -> no predication; EXEC stays all-ones as WMMA requires.
template<int DOUT, bool RELU, bool SCALE_OUT>
__global__ __launch_bounds__(32 * (DOUT / 16))
void gemm_wmma_kernel(const float* __restrict__ A,       // [N, FEAT] aggregated
                      const float* __restrict__ normd,   // [N] pre-scale (norm_dst)
                      const float* __restrict__ norms,   // [N] post-scale (norm_src), may be null
                      const float* __restrict__ W,       // [FEAT, DOUT] row-major
                      const float* __restrict__ bias,    // [DOUT]
                      float* __restrict__ out) {         // [N, DOUT]
    constexpr int K = FEAT;
    const int lane = threadIdx.x & 31;
    const int n0   = (threadIdx.x >> 5) << 4;    // column tile from wave id
    const int m0   = blockIdx.x << 4;            // row tile
    const int li   = lane & 15;                  // M (for A) / N (for B) within tile
    const int kh   = (lane >> 4) << 1;           // K sub-offset: lanes 0-15 -> {K0,K1}, 16-31 -> {K2,K3}

    const int row = m0 + li;
    const float nd = normd[row];
    const float* __restrict__ arow = A + (size_t)row * K;

    v8f acc = {};
#pragma unroll
    for (int k0 = 0; k0 < K; k0 += 4) {
        v2f a = *(const v2f*)(arow + k0 + kh);   // 8B-aligned: k0+kh is even
        a *= nd;                                  // fuse D_in^{-1/2}
        v2f b;
        b.x = W[(size_t)(k0 + kh)     * DOUT + n0 + li];
        b.y = W[(size_t)(k0 + kh + 1) * DOUT + n0 + li];
        acc = __builtin_amdgcn_wmma_f32_16x16x4_f32(false, a, false, b,
                                                    (short)0, acc, false, false);
    }

    // Epilogue: C/D layout -> VGPR v holds (M = m0 + 8*(lane>>4) + v, N = n0 + li)
    const int col   = n0 + li;
    const float bv  = bias[col];
    const int mbase = m0 + ((lane >> 4) << 3);
#pragma unroll
    for (int v = 0; v < 8; ++v) {
        float x = acc[v] + bv;
        if (RELU) x = fmaxf(x, 0.0f);
        if (SCALE_OUT) x *= norms[mbase + v];     // fuse next layer's D_out^{-1/2}
        out[(size_t)(mbase + v) * DOUT + col] = x;
    }
}

// ---------------- mean readout per graph ----------------
__global__ void readout_sum_kernel(const float* __restrict__ h, const int* __restrict__ gid,
                                   float* __restrict__ gsum, float* __restrict__ gcnt, int N) {
    int t = blockIdx.x * blockDim.x + threadIdx.x;   // N*32 threads
    int n = t >> 5;
    if (n >= N) return;
    int c = t & 31;
    int g = gid[n];
    atomicAdd(&gsum[g * NCLS + c], h[(size_t)n * NCLS + c]);
    if (c == 0) atomicAdd(&gcnt[g], 1.0f);
}

__global__ void readout_div_kernel(const float* __restrict__ gsum, const float* __restrict__ gcnt,
                                   float* __restrict__ out) {
    int t = blockIdx.x * blockDim.x + threadIdx.x;
    if (t < N_GRAPHS * NCLS)
        out[t] = gsum[t] / fmaxf(gcnt[t >> 5], 1.0f);
}

// ---------------- driver ----------------
extern "C" void kernel_launch(void* const* d_in, const int* in_sizes, int n_in,
                              void* d_out, int out_size, void* d_ws, size_t ws_size,
                              hipStream_t stream) {
    const float* in_feat = (const float*)d_in[0];
    const int*   src     = (const int*)  d_in[1];
    const int*   dst     = (const int*)  d_in[2];
    const int*   gid     = (const int*)  d_in[3];
    const float* W1      = (const float*)d_in[4];
    const float* b1      = (const float*)d_in[5];
    const float* W2      = (const float*)d_in[6];
    const float* b2      = (const float*)d_in[7];
    const float* W3      = (const float*)d_in[8];
    const float* b3      = (const float*)d_in[9];
    (void)in_sizes; (void)n_in; (void)out_size; (void)ws_size;

    float* ws       = (float*)d_ws;
    float* deg_out  = ws;
    float* deg_in   = ws + (size_t)N_NODES;
    float* norm_src = ws + (size_t)2 * N_NODES;
    float* norm_dst = ws + (size_t)3 * N_NODES;
    float* gsum     = ws + (size_t)4 * N_NODES;
    float* gcnt     = gsum + N_GRAPHS * NCLS;
    float* buf0     = ws + (size_t)4 * N_NODES + 4096;   // [N, FEAT] ping (16B aligned)
    float* buf1     = buf0 + (size_t)N_NODES * FEAT;     // [N, FEAT] pong

    const size_t rowBytes = (size_t)N_NODES * FEAT * sizeof(float);
    const int eGrid  = (N_EDGES * 32) / 256;             // exact: 200000
    const int nGrid  = (N_NODES * 32 + 255) / 256;
    const int mTiles = N_NODES / 16;                     // exact: 3125

    // degrees + norms
    hipMemsetAsync(deg_out, 0, (size_t)2 * N_NODES * sizeof(float), stream);
    hipMemsetAsync(gsum, 0, (N_GRAPHS * NCLS + N_GRAPHS) * sizeof(float), stream);
    degrees_kernel<<<(N_EDGES + 255) / 256, 256, 0, stream>>>(src, dst, deg_out, deg_in, N_EDGES);
    norm_kernel<<<(N_NODES + 255) / 256, 256, 0, stream>>>(deg_out, deg_in, norm_src, norm_dst, N_NODES);

    // layer 1: xs = in_feat * norm_src; agg; h1s = relu(agg*nd @ W1 + b1) * norm_src
    scale_rows_kernel<<<nGrid, 256, 0, stream>>>(in_feat, norm_src, buf0, N_NODES);
    hipMemsetAsync(buf1, 0, rowBytes, stream);
    edge_agg_kernel<<<eGrid, 256, 0, stream>>>(buf0, src, dst, buf1, N_EDGES);
    gemm_wmma_kernel<FEAT, true, true><<<mTiles, 256, 0, stream>>>(buf1, norm_dst, norm_src, W1, b1, buf0);

    // layer 2
    hipMemsetAsync(buf1, 0, rowBytes, stream);
    edge_agg_kernel<<<eGrid, 256, 0, stream>>>(buf0, src, dst, buf1, N_EDGES);
    gemm_wmma_kernel<FEAT, true, true><<<mTiles, 256, 0, stream>>>(buf1, norm_dst, norm_src, W2, b2, buf0);

    // layer 3 (no relu, no post-scale) -> buf0 as dense [N, NCLS]
    hipMemsetAsync(buf1, 0, rowBytes, stream);
    edge_agg_kernel<<<eGrid, 256, 0, stream>>>(buf0, src, dst, buf1, N_EDGES);
    gemm_wmma_kernel<NCLS, false, false><<<mTiles, 64, 0, stream>>>(buf1, norm_dst, nullptr, W3, b3, buf0);

    // mean readout
    readout_sum_kernel<<<nGrid, 256, 0, stream>>>(buf0, gid, gsum, gcnt, N_NODES);
    readout_div_kernel<<<(N_GRAPHS * NCLS + 255) / 256, 256, 0, stream>>>(gsum, gcnt, (float*)d_out);
}